// conv3DLookupLayer_18373870092829
// MI455X (gfx1250) — compile-verified
//
#include <hip/hip_runtime.h>

// Sizes from the reference
#define NB      16      // batch
#define NQ      512     // query dim
#define CIN     32
#define COUT    32
#define KS      3
#define DIM     32      // D = H = W
#define DOUT    30      // DIM - KS + 1
#define KTOT    864     // CIN * 27
#define NCOL    27648   // K^3 * CIN * COUT  (lut columns)
#define NTILES  1728    // NCOL / 16
#define KHROWS  4       // kh rows staged in LDS (2 ho rows x 3-tap window)

typedef __attribute__((ext_vector_type(2))) float v2f;
typedef __attribute__((ext_vector_type(8))) float v8f;

// -----------------------------------------------------------------------------
// Kernel A: weights = query @ lut, stored permuted as W2[b][cout][kd*kh*kw][cin]
// WMMA f32 16x16x4: M = 16 batches, K = 512, N = 16 lut columns per wave-tile.
// Bandwidth-bound: streams the 56.6 MB lut exactly once (prefetch ahead).
// -----------------------------------------------------------------------------
__global__ __launch_bounds__(256) void build_weights_kernel(
    const float* __restrict__ query,   // [16][512]
    const float* __restrict__ lut,     // [512][27648]
    float* __restrict__ w2)            // [16][COUT][27][CIN]
{
  const int lane = threadIdx.x & 31;
  const int wave = threadIdx.x >> 5;
  const int tile = blockIdx.x * 8 + wave;      // grid sized exactly: 216*8 = 1728
  const int n0   = tile * 16;
  const int half = lane >> 4;                  // 0 = lanes 0-15, 1 = lanes 16-31
  const int l15  = lane & 15;

  const float* qbase = query + l15 * NQ + half * 2;                 // A: row = batch
  const float* lbase = lut + (size_t)(half * 2) * NCOL + n0 + l15;  // B: column

  v8f acc = {};
  for (int q = 0; q < NQ; q += 4) {
    __builtin_prefetch(lbase + (size_t)(q + 64) * NCOL, 0, 1);
    v2f a;
    a.x = qbase[q];
    a.y = qbase[q + 1];
    v2f b;
    b.x = lbase[(size_t)q * NCOL];
    b.y = lbase[(size_t)(q + 1) * NCOL];
    acc = __builtin_amdgcn_wmma_f32_16x16x4_f32(
        /*neg_a=*/false, a, /*neg_b=*/false, b,
        /*c_mod=*/(short)0, acc, /*reuse_a=*/false, /*reuse_b=*/false);
  }

  // Scatter-store with permutation: n -> (cout, cin, kdhw); dest K-order = kdhw*32 + cin
  const int n    = n0 + l15;
  const int kdhw = n % 27;
  const int rest = n / 27;          // cout*32 + cin
  const int cin  = rest & 31;
  const int cout = rest >> 5;
  const size_t off = (size_t)cout * KTOT + (size_t)kdhw * CIN + cin;
#pragma unroll
  for (int r = 0; r < 8; ++r) {
    const int b = r + half * 8;     // D layout: vgpr r = rows r / r+8
    w2[(size_t)b * (COUT * KTOT) + off] = acc[r];
  }
}

// -----------------------------------------------------------------------------
// Kernel B: per-batch conv3d as implicit GEMM.
// Block = (b, d0, ho-pair); 4 waves = {cout tile 0/16} x {ho row 0/1}.
// Each wave keeps TWO accumulators (wo tiles @0 and @14) that SHARE the A
// operand: per 2 wmma -> 1 global b64 (A) + 2 ds b64 (B).
// LDS slab[kd][khh(4)][w][cin] = 48 KB, cin innermost -> aligned ds_load_b64.
// 432 wmma f32 16x16x4 per wave, two independent accumulation chains.
// -----------------------------------------------------------------------------
__global__ __launch_bounds__(128) void conv_wmma_kernel(
    const float* __restrict__ target,  // [16][32][32][32][32]
    const float* __restrict__ w2,      // [16][COUT][27][CIN]
    float* __restrict__ out)           // [16][32][30][30][30]
{
  __shared__ float slab[KS * KHROWS * DIM * CIN];   // 3*4*32*32 floats = 48 KB

  const int bidx = blockIdx.x;                      // b*450 + d0*15 + hp
  const int b    = bidx / (DOUT * (DOUT / 2));
  const int rem  = bidx % (DOUT * (DOUT / 2));
  const int d0   = rem / (DOUT / 2);
  const int h02  = (rem % (DOUT / 2)) * 2;          // first of two ho rows

  // ---- cooperative LDS fill: global-coalesced reads (w fastest) -------------
  {
    const float* tb = target + (size_t)b * CIN * DIM * DIM * DIM;
    // i over (kd, khh, cin, w): 3*4*32*32 = 12288 elements, 96 per thread
    for (int i = threadIdx.x; i < KS * KHROWS * CIN * DIM; i += 128) {
      const int w   = i & 31;
      const int cin = (i >> 5) & 31;
      const int t   = i >> 10;            // kd*4 + khh, 0..11
      const int kd  = t >> 2;
      const int khh = t & 3;
      const float v = tb[(size_t)cin * (DIM * DIM * DIM)
                         + (size_t)(d0 + kd) * (DIM * DIM)
                         + (size_t)(h02 + khh) * DIM + w];
      slab[((t * DIM) + w) * CIN + cin] = v;
    }
  }
  __syncthreads();

  const int lane  = threadIdx.x & 31;
  const int wave  = threadIdx.x >> 5;
  const int mtile = wave & 1;             // cout half
  const int hosel = wave >> 1;            // which of the two ho rows
  const int cout0 = mtile * 16;
  const int ho    = h02 + hosel;
  const int half  = lane >> 4;
  const int l15   = lane & 15;

  // A: W2[b][cout0 + l15][ K = kk*32 + c0 + half*2 + {0,1} ]
  const float* abase = w2 + (size_t)b * (COUT * KTOT)
                          + (size_t)(cout0 + l15) * KTOT + half * 2;

  v8f acc0 = {};                          // wo tile [0..15]
  v8f acc1 = {};                          // wo tile [14..29]
#pragma unroll
  for (int kd = 0; kd < KS; ++kd) {
#pragma unroll
    for (int kh = 0; kh < KS; ++kh) {
      const int khh = hosel + kh;         // slab kh row for this wave's ho
#pragma unroll
      for (int kw = 0; kw < KS; ++kw) {
        const int kk = (kd * 3 + kh) * 3 + kw;
        const float* arow  = abase + kk * CIN;
        const float* brow0 = &slab[(((kd * KHROWS + khh) * DIM) + (l15 + kw)) * CIN
                                   + half * 2];
        const float* brow1 = brow0 + 14 * CIN;
#pragma unroll
        for (int c0 = 0; c0 < CIN; c0 += 4) {
          v2f a;
          a.x = arow[c0];
          a.y = arow[c0 + 1];
          v2f b0;
          b0.x = brow0[c0];
          b0.y = brow0[c0 + 1];
          v2f b1;
          b1.x = brow1[c0];
          b1.y = brow1[c0 + 1];
          acc0 = __builtin_amdgcn_wmma_f32_16x16x4_f32(
              false, a, false, b0, (short)0, acc0, false, false);
          acc1 = __builtin_amdgcn_wmma_f32_16x16x4_f32(
              false, a, false, b1, (short)0, acc1, false, false);
        }
      }
    }
  }

  // ---- store: each output written exactly once ------------------------------
  // wo tile @14 recomputes wo=14,15 (EXEC stays all-ones for WMMA) but only
  // tile @0 stores them.
  {
    const size_t obase = ((((size_t)b * COUT + cout0 + half * 8) * DOUT + d0) * DOUT + ho)
                         * DOUT;
#pragma unroll
    for (int r = 0; r < 8; ++r) {
      out[obase + (size_t)r * (DOUT * DOUT * DOUT) + l15] = acc0[r];
    }
    if (l15 >= 2) {
#pragma unroll
      for (int r = 0; r < 8; ++r) {
        out[obase + (size_t)r * (DOUT * DOUT * DOUT) + 14 + l15] = acc1[r];
      }
    }
  }
}

extern "C" void kernel_launch(void* const* d_in, const int* in_sizes, int n_in,
                              void* d_out, int out_size, void* d_ws, size_t ws_size,
                              hipStream_t stream) {
  const float* query  = (const float*)d_in[0];   // [16, 512]
  const float* target = (const float*)d_in[1];   // [16, 32, 32, 32, 32]
  const float* lut    = (const float*)d_in[2];   // [512, 27648]
  float* out = (float*)d_out;                    // [16, 32, 30, 30, 30]
  float* w2  = (float*)d_ws;                     // needs 16*27648*4 = 1.77 MB scratch

  // Kernel A: 1728 column tiles, 8 wave-tiles per 256-thread block
  build_weights_kernel<<<NTILES / 8, 256, 0, stream>>>(query, lut, w2);

  // Kernel B: one block per (b, d0, ho-pair); 4 waves per block
  conv_wmma_kernel<<<NB * DOUT * (DOUT / 2), 128, 0, stream>>>(target, w2, out);
}